// BaseNeuralNetwork_2671469658783
// MI455X (gfx1250) — compile-verified
//
#include <hip/hip_runtime.h>
#include <hip/hip_bf16.h>
#include <math.h>

#define SEQ   2048
#define BATCH 64
#define D     256
#define N1    100
#define N2    150

typedef __attribute__((ext_vector_type(2))) float v2f;
typedef __attribute__((ext_vector_type(8))) float v8f;

// ---------------------------------------------------------------------------
// Kernel 1: per-row tf -> scores -> tf-idf weighted embedding pooling.
// grid = BATCH blocks, 256 threads. LDS: 2048 tokens + 2048 scores (16 KB).
// ---------------------------------------------------------------------------
__global__ void tfidf_pool_kernel(const int* __restrict__ x,
                                  const float* __restrict__ emb,
                                  const float* __restrict__ idf,
                                  float* __restrict__ pooled) {
    __shared__ int   s_tok[SEQ];
    __shared__ float s_sc[SEQ];
    const int b = blockIdx.x;
    const int t = threadIdx.x;

    // tokens[b][s] = x[s*BATCH + b]   (x is [SEQ, BATCH])
    for (int s = t; s < SEQ; s += 256) s_tok[s] = x[s * BATCH + b];
    __syncthreads();

    // tf via match count over the row; score = (tok==0) ? 0 : tf * idf[tok]
    for (int s = t; s < SEQ; s += 256) {
        const int tok = s_tok[s];
        int cnt = 0;
        for (int j = 0; j < SEQ; ++j) cnt += (s_tok[j] == tok) ? 1 : 0;
        s_sc[s] = (tok == 0) ? 0.0f : (float)cnt * idf[tok];
    }
    __syncthreads();

    // pooled[b][d] = sum_s score[s] * emb[tok[s]][d]; thread t owns dim d=t.
    const int d = t;
    float acc = 0.0f;
    for (int s = 0; s < SEQ; ++s) {
        const float sc = s_sc[s];
        if (sc != 0.0f) acc += sc * emb[s_tok[s] * D + d];
    }
    pooled[b * D + d] = acc;
}

// ---------------------------------------------------------------------------
// f32 WMMA 16x16 tile GEMM: D = A(MxK) * W^T(KxN), K-loop in steps of 4 with
// V_WMMA_F32_16X16X4_F32.  W row-major [N x K] so B[k][n] = W[n*ldw + k].
//
// Fragment mapping (wave32, ISA 7.12.2):
//   A:   lane lr in [0,16), row = m0+lr; half = lane>>4; a = {A[., k0+2h], A[., k0+2h+1]}
//   B:   lane lr -> col = n0+lr;                         b = {B[k0+2h][.], B[k0+2h+1][.]}
//   C/D: VGPR v -> row m0 + v + 8h, col n0 + lr
//
// Out-of-range columns (col >= N): WMMA is column-separable, so garbage in B
// column n only pollutes D column n, which the (guarded) store discards.
// We therefore only CLAMP the address in-bounds -- no zero-masking, no
// exec-divergent loads in the inner loop.
// ---------------------------------------------------------------------------
__device__ __forceinline__ v8f gemm_tile_f32(const float* __restrict__ A, int lda,
                                             const float* __restrict__ W, int ldw,
                                             int N, int K, int m0, int n0,
                                             int half, int lr) {
    v8f acc = {0.f, 0.f, 0.f, 0.f, 0.f, 0.f, 0.f, 0.f};
    const int row = m0 + lr;                       // always < 64 here
    const int col = n0 + lr;
    const int cc  = (col < N) ? col : (N - 1);     // clamp: stay in-bounds
    const float* __restrict__ Arow = A + row * lda + 2 * half;
    const float* __restrict__ Wrow = W + cc  * ldw + 2 * half;
    for (int k0 = 0; k0 < K; k0 += 4) {
        v2f a, bfr;
        a.x   = Arow[k0];
        a.y   = Arow[k0 + 1];
        bfr.x = Wrow[k0];
        bfr.y = Wrow[k0 + 1];
        acc = __builtin_amdgcn_wmma_f32_16x16x4_f32(
            /*neg_a=*/false, a, /*neg_b=*/false, bfr,
            /*c_mod=*/(short)0, acc, /*reuse_a=*/false, /*reuse_b=*/false);
    }
    return acc;
}

__device__ __forceinline__ void store_tile_bias_relu(v8f acc, float* __restrict__ H,
                                                     int ldH, int N, int m0, int n0,
                                                     int half, int lr,
                                                     const float* __restrict__ bias) {
    const int col = n0 + lr;
    if (col < N) {
        const float bb = bias[col];
        #pragma unroll
        for (int v = 0; v < 8; ++v) {
            const int r = m0 + v + 8 * half;
            H[r * ldH + col] = fmaxf(acc[v] + bb, 0.0f);
        }
    }
}

// ---------------------------------------------------------------------------
// Kernel 2: full MLP head in one workgroup (8 waves, 256 threads).
//   L1: pooled[64x256] @ W1^T -> relu -> h1[64x100]   (28 tiles, K=256)
//   L2: h1[64x100]     @ W2^T -> relu -> h2[64x150]   (40 tiles, K=100)
//   L3: h2[64x150]     @ W3^T + b3 -> softmax -> out[64x2]  (scalar, 64 lanes)
// Tile loops are wave-uniform, so EXEC is all-1s at every WMMA.
// ---------------------------------------------------------------------------
__global__ void mlp_wmma_kernel(const float* __restrict__ pooled,
                                const float* __restrict__ W1, const float* __restrict__ b1,
                                const float* __restrict__ W2, const float* __restrict__ b2,
                                const float* __restrict__ W3, const float* __restrict__ b3,
                                float* __restrict__ h1, float* __restrict__ h2,
                                float* __restrict__ out) {
    const int lane = threadIdx.x & 31;
    const int wave = threadIdx.x >> 5;   // 0..7
    const int half = lane >> 4;
    const int lr   = lane & 15;

    // ---- Layer 1: M=64, N=100, K=256 -> 4 x 7 = 28 tiles
    {
        const int nTilesN = (N1 + 15) / 16;              // 7
        const int nTiles  = 4 * nTilesN;                 // 28
        for (int tile = wave; tile < nTiles; tile += 8) {
            const int m0 = (tile / nTilesN) * 16;
            const int n0 = (tile % nTilesN) * 16;
            v8f acc = gemm_tile_f32(pooled, D, W1, D, N1, D, m0, n0, half, lr);
            store_tile_bias_relu(acc, h1, N1, N1, m0, n0, half, lr, b1);
        }
    }
    __syncthreads();

    // ---- Layer 2: M=64, N=150, K=100 -> 4 x 10 = 40 tiles
    {
        const int nTilesN = (N2 + 15) / 16;              // 10
        const int nTiles  = 4 * nTilesN;                 // 40
        for (int tile = wave; tile < nTiles; tile += 8) {
            const int m0 = (tile / nTilesN) * 16;
            const int n0 = (tile % nTilesN) * 16;
            v8f acc = gemm_tile_f32(h1, N1, W2, N1, N2, N1, m0, n0, half, lr);
            store_tile_bias_relu(acc, h2, N2, N2, m0, n0, half, lr, b2);
        }
    }
    __syncthreads();

    // ---- Layer 3 + softmax (64 lanes, scalar dot over K=150)
    if (threadIdx.x < BATCH) {
        const int b = threadIdx.x;
        float l0 = b3[0], l1 = b3[1];
        for (int k = 0; k < N2; ++k) {
            const float h = h2[b * N2 + k];
            l0 += h * W3[0 * N2 + k];
            l1 += h * W3[1 * N2 + k];
        }
        const float m  = fmaxf(l0, l1);
        const float e0 = __expf(l0 - m);
        const float e1 = __expf(l1 - m);
        const float inv = 1.0f / (e0 + e1);
        out[b * 2 + 0] = e0 * inv;
        out[b * 2 + 1] = e1 * inv;
    }
}

// ---------------------------------------------------------------------------
extern "C" void kernel_launch(void* const* d_in, const int* in_sizes, int n_in,
                              void* d_out, int out_size, void* d_ws, size_t ws_size,
                              hipStream_t stream) {
    const int*   x   = (const int*)  d_in[0];
    const float* emb = (const float*)d_in[1];
    const float* idf = (const float*)d_in[2];
    const float* W1  = (const float*)d_in[3];
    const float* b1  = (const float*)d_in[4];
    const float* W2  = (const float*)d_in[5];
    const float* b2  = (const float*)d_in[6];
    const float* W3  = (const float*)d_in[7];
    const float* b3  = (const float*)d_in[8];
    float*       out = (float*)d_out;

    float* pooled = (float*)d_ws;              // 64*256 floats
    float* h1     = pooled + BATCH * D;        // 64*100 floats
    float* h2     = h1 + BATCH * N1;           // 64*150 floats

    tfidf_pool_kernel<<<BATCH, 256, 0, stream>>>(x, emb, idf, pooled);
    mlp_wmma_kernel<<<1, 256, 0, stream>>>(pooled, W1, b1, W2, b2, W3, b3, h1, h2, out);
}